// GaussianSplattingData_35794257445477
// MI455X (gfx1250) — compile-verified
//
#include <hip/hip_runtime.h>
#include <hip/hip_bf16.h>

typedef __attribute__((ext_vector_type(2))) float v2f;
typedef __attribute__((ext_vector_type(8))) float v8f;

#define NG      1024
#define IMG_W   128
#define IMG_H   128
#define NPIX    (IMG_W * IMG_H)
#define EPS2D   0.3f
#define ACLAMP  0.999f
#define AMIN    (1.0f / 255.0f)
#define NEARZ   0.01f

// workspace layout (floats):
//   [0,        NG*8)  : sorted gaussian sigma-features, [g][8] = {g0..g5,0,0}
//   [NG*8, NG*8+NG*4) : sorted color4, [g][4] = {r,g,b,opac}

__global__ void __launch_bounds__(NG)
gs_prep(const float* __restrict__ means, const float* __restrict__ quats,
        const float* __restrict__ scales, const float* __restrict__ opac,
        const float* __restrict__ colors, const float* __restrict__ vm,
        const float* __restrict__ Kc, float* __restrict__ ws)
{
    extern __shared__ float lds[];
    float* ufeat = lds;                 // NG*8
    float* ucol  = lds + NG * 8;        // NG*4
    float* key   = lds + NG * 12;       // NG
    int*   sidx  = (int*)(lds + NG * 13); // NG

    const int i = threadIdx.x;

    const float R00 = vm[0], R01 = vm[1], R02 = vm[2],  t0 = vm[3];
    const float R10 = vm[4], R11 = vm[5], R12 = vm[6],  t1 = vm[7];
    const float R20 = vm[8], R21 = vm[9], R22 = vm[10], t2 = vm[11];
    const float fx = Kc[0], cx = Kc[2], fy = Kc[4], cy = Kc[5];

    const float mX = means[i*3+0], mY = means[i*3+1], mZ = means[i*3+2];
    const float x = R00*mX + R01*mY + R02*mZ + t0;
    const float y = R10*mX + R11*mY + R12*mZ + t1;
    const float z = R20*mX + R21*mY + R22*mZ + t2;
    const bool valid = z > NEARZ;
    const float zc   = valid ? z : 1.0f;
    const float invz = 1.0f / zc;

    // quaternion -> rotation, scaled columns
    float qw = quats[i*4+0], qx = quats[i*4+1], qy = quats[i*4+2], qz = quats[i*4+3];
    const float qn = rsqrtf(qw*qw + qx*qx + qy*qy + qz*qz);
    qw *= qn; qx *= qn; qy *= qn; qz *= qn;
    const float s0 = __expf(scales[i*3+0]);
    const float s1 = __expf(scales[i*3+1]);
    const float s2 = __expf(scales[i*3+2]);
    const float M00 = (1.f - 2.f*(qy*qy + qz*qz)) * s0;
    const float M01 = (2.f*(qx*qy - qw*qz)) * s1;
    const float M02 = (2.f*(qx*qz + qw*qy)) * s2;
    const float M10 = (2.f*(qx*qy + qw*qz)) * s0;
    const float M11 = (1.f - 2.f*(qx*qx + qz*qz)) * s1;
    const float M12 = (2.f*(qy*qz - qw*qx)) * s2;
    const float M20 = (2.f*(qx*qz - qw*qy)) * s0;
    const float M21 = (2.f*(qy*qz + qw*qx)) * s1;
    const float M22 = (1.f - 2.f*(qx*qx + qy*qy)) * s2;

    // cov3 = M M^T (symmetric)
    const float c00 = M00*M00 + M01*M01 + M02*M02;
    const float c01 = M00*M10 + M01*M11 + M02*M12;
    const float c02 = M00*M20 + M01*M21 + M02*M22;
    const float c11 = M10*M10 + M11*M11 + M12*M12;
    const float c12 = M10*M20 + M11*M21 + M12*M22;
    const float c22 = M20*M20 + M21*M21 + M22*M22;

    // covc = R cov3 R^T
    const float A00 = R00*c00 + R01*c01 + R02*c02;
    const float A01 = R00*c01 + R01*c11 + R02*c12;
    const float A02 = R00*c02 + R01*c12 + R02*c22;
    const float A10 = R10*c00 + R11*c01 + R12*c02;
    const float A11 = R10*c01 + R11*c11 + R12*c12;
    const float A12 = R10*c02 + R11*c12 + R12*c22;
    const float A20 = R20*c00 + R21*c01 + R22*c02;
    const float A21 = R20*c01 + R21*c11 + R22*c12;
    const float A22 = R20*c02 + R21*c12 + R22*c22;
    const float cc00 = A00*R00 + A01*R01 + A02*R02;
    const float cc01 = A00*R10 + A01*R11 + A02*R12;
    const float cc02 = A00*R20 + A01*R21 + A02*R22;
    const float cc11 = A10*R10 + A11*R11 + A12*R12;
    const float cc12 = A10*R20 + A11*R21 + A12*R22;
    const float cc22 = A20*R20 + A21*R21 + A22*R22;

    // perspective Jacobian, 2D covariance, conic
    const float j0 = fx*invz, j2 = -fx*x*invz*invz;
    const float j4 = fy*invz, j5 = -fy*y*invz*invz;
    const float a = j0*j0*cc00 + 2.f*j0*j2*cc02 + j2*j2*cc22 + EPS2D;
    const float b = j0*(cc01*j4 + cc02*j5) + j2*(cc12*j4 + cc22*j5);
    const float c = j4*j4*cc11 + 2.f*j4*j5*cc12 + j5*j5*cc22 + EPS2D;
    const float det = a*c - b*b;
    const float ia  =  c / det;
    const float ibv = -b / det;
    const float icv =  a / det;
    const float pmx = fx*x*invz + cx;
    const float pmy = fy*y*invz + cy;
    const float o = 1.0f / (1.0f + __expf(-opac[i]));

    // rank-6 feature vector: sigma = f(pixel) . g(gaussian)
    float g0, g1, g2, g3, g4, g5;
    if (valid) {
        g0 = 0.5f*ia; g1 = 0.5f*icv; g2 = ibv;
        g3 = -(ia*pmx + ibv*pmy);
        g4 = -(icv*pmy + ibv*pmx);
        g5 = 0.5f*ia*pmx*pmx + 0.5f*icv*pmy*pmy + ibv*pmx*pmy;
    } else {
        g0 = g1 = g2 = g3 = g4 = 0.f;
        g5 = 1e30f;   // sigma -> +inf -> alpha == 0, matches valid-mask
    }
    ufeat[i*8+0] = g0; ufeat[i*8+1] = g1; ufeat[i*8+2] = g2;
    ufeat[i*8+3] = g3; ufeat[i*8+4] = g4; ufeat[i*8+5] = g5;
    ufeat[i*8+6] = 0.f; ufeat[i*8+7] = 0.f;
    ucol[i*4+0] = colors[i*3+0];
    ucol[i*4+1] = colors[i*3+1];
    ucol[i*4+2] = colors[i*3+2];
    ucol[i*4+3] = o;
    key[i]  = z;
    sidx[i] = i;
    __syncthreads();

    // bitonic sort (ascending z) of 1024 (key, idx) pairs
    for (int k = 2; k <= NG; k <<= 1) {
        for (int j = k >> 1; j > 0; j >>= 1) {
            const int ixj = i ^ j;
            if (ixj > i) {
                const float ki = key[i], kj = key[ixj];
                const bool asc = ((i & k) == 0);
                if ((ki > kj) == asc) {
                    key[i] = kj; key[ixj] = ki;
                    const int t = sidx[i]; sidx[i] = sidx[ixj]; sidx[ixj] = t;
                }
            }
            __syncthreads();
        }
    }

    // gather into sorted order and write to workspace
    const int s = sidx[i];
    #pragma unroll
    for (int t = 0; t < 8; ++t) ws[i*8 + t] = ufeat[s*8 + t];
    #pragma unroll
    for (int t = 0; t < 4; ++t) ws[NG*8 + i*4 + t] = ucol[s*4 + t];
}

// Each wave renders 16 pixels. Per 16-gaussian chunk:
//   D(16 gauss x 16 pix) = A(gauss feats 16x8) x B(pixel feats 8x16) via two
//   chained V_WMMA_F32_16X16X4_F32; then per-lane serial composite over the
//   8 sorted gaussians this lane holds, one cross-half shuffle stitches order.
__global__ void __launch_bounds__(256)
gs_render(const float* __restrict__ ws, float* __restrict__ out)
{
    extern __shared__ float lds[];
    float* sg = lds;            // NG*8 gaussian features
    float* sc = lds + NG * 8;   // NG*4 color4

    // Stage the 48 KB sorted gaussian table cache->LDS with the CDNA5 async
    // path (no VGPR round-trip). Dynamic-LDS allocation starts at offset 0,
    // so VDST holds the plain byte offset (HW adds LDS_BASE).
    {
        const unsigned tid = threadIdx.x;
        #pragma unroll
        for (int k = 0; k < (NG * 12 * 4) / 16 / 256; ++k) {   // 12 iters
            const unsigned byteOff = (tid + (unsigned)k * 256u) * 16u;
            const char* src = (const char*)ws + byteOff;
            asm volatile("global_load_async_to_lds_b128 %0, %1, off"
                         :: "v"(byteOff), "v"(src)
                         : "memory");
        }
        asm volatile("s_wait_asynccnt 0x0" ::: "memory");
    }
    __syncthreads();

    const int lane = threadIdx.x & 31;
    const int n    = lane & 15;       // N index (pixel) and A-row (gaussian)
    const int h    = lane >> 4;       // half-wave
    const int gw   = blockIdx.x * (blockDim.x >> 5) + (threadIdx.x >> 5);
    const int pix  = gw * 16 + n;
    const float px = (float)(pix % IMG_W) + 0.5f;
    const float py = (float)(pix / IMG_W) + 0.5f;

    // B operand (pixel features), K = 2h+v per ISA layout
    v2f b1, b2;
    if (h == 0) { b1[0] = px*px; b1[1] = py*py; b2[0] = py;  b2[1] = 1.f; }
    else        { b1[0] = px*py; b1[1] = px;    b2[0] = 0.f; b2[1] = 0.f; }

    float T = 1.0f;
    float accR = 0.f, accG = 0.f, accB = 0.f, accA = 0.f;

    for (int cblk = 0; cblk < NG / 16; ++cblk) {
        const int gbase = cblk * 16;
        // A operand: this lane supplies gaussian (gbase+n), features K=2h..2h+1 / 4+2h..5+2h
        const v2f a1 = *reinterpret_cast<const v2f*>(sg + (gbase + n) * 8 + 2*h);
        const v2f a2 = *reinterpret_cast<const v2f*>(sg + (gbase + n) * 8 + 4 + 2*h);

        v8f sig = {};
        sig = __builtin_amdgcn_wmma_f32_16x16x4_f32(false, a1, false, b1,
                                                    (short)0, sig, false, false);
        sig = __builtin_amdgcn_wmma_f32_16x16x4_f32(false, a2, false, b2,
                                                    (short)0, sig, false, false);

        // lane holds sigma for gaussians gbase + 8h + r (r=0..7), pixel = pix
        float Tl = 1.0f, cR = 0.f, cG = 0.f, cB = 0.f, cA = 0.f;
        #pragma unroll
        for (int r = 0; r < 8; ++r) {
            const float4 col = reinterpret_cast<const float4*>(sc)[gbase + 8*h + r];
            const float sv = sig[r];
            float al = fminf(col.w * __expf(-sv), ACLAMP);
            al = (sv >= 0.f && al >= AMIN) ? al : 0.f;
            const float w = Tl * al;
            cR += w * col.x; cG += w * col.y; cB += w * col.z; cA += w;
            Tl *= (1.f - al);
        }
        // stitch the two sorted halves: back half scales by front half's product
        const float Pother = __shfl_xor(Tl, 16, 32);
        const float scale  = h ? (T * Pother) : T;
        accR += scale * cR; accG += scale * cG;
        accB += scale * cB; accA += scale * cA;
        T *= Tl * Pother;
    }

    // combine partial sums from both halves (same pixel lives in lane n and n+16)
    accR += __shfl_xor(accR, 16, 32);
    accG += __shfl_xor(accG, 16, 32);
    accB += __shfl_xor(accB, 16, 32);
    accA += __shfl_xor(accA, 16, 32);

    if (h == 0) {
        out[pix*3 + 0] = accR;
        out[pix*3 + 1] = accG;
        out[pix*3 + 2] = accB;
        out[NPIX*3 + pix] = accA;
    }
}

extern "C" void kernel_launch(void* const* d_in, const int* in_sizes, int n_in,
                              void* d_out, int out_size, void* d_ws, size_t ws_size,
                              hipStream_t stream)
{
    const float* means  = (const float*)d_in[0];
    const float* quats  = (const float*)d_in[1];
    const float* scales = (const float*)d_in[2];
    const float* opacs  = (const float*)d_in[3];
    const float* colors = (const float*)d_in[4];
    const float* vms    = (const float*)d_in[5];
    const float* Ks     = (const float*)d_in[6];
    float* ws  = (float*)d_ws;
    float* out = (float*)d_out;

    // stage 1: project + sort (one 1024-thread workgroup, 56 KB LDS)
    gs_prep<<<1, NG, (size_t)(NG * 14) * sizeof(float), stream>>>(
        means, quats, scales, opacs, colors, vms, Ks, ws);

    // stage 2: WMMA tiled render; 1024 waves x 16 pixels = 16384 pixels
    const int blocks = (NPIX / 16) / 8;   // 8 waves of 32 per block
    gs_render<<<blocks, 256, (size_t)(NG * 12) * sizeof(float), stream>>>(ws, out);
}